// RoIAlign_16527034155028
// MI455X (gfx1250) — compile-verified
//
#include <hip/hip_runtime.h>

// RoIAlign (TF crop_and_resize) for MI455X (gfx1250).
//
// Roofline: output = 2000*256*14*14 f32 = 401 MB streamed once (floor ~17 us
// at 23.3 TB/s HBM); features = 64 MB, fully resident in the 192 MB L2 and
// gathered from there. Arithmetic intensity ~0.4 FLOP/B -> pure memory bound.
// WMMA does not apply (densifying the 4-nonzero/column interp matrix would
// blow up L2 traffic ~30x); TDM cannot stage the data-dependent footprints
// (worst case 1 MB per 16-channel tile > 320 KB LDS). Optimal path = direct
// L2 gather + non-temporal stores (TH_STORE_NT) so the output stream does not
// evict the L2-resident feature map.
//
// wave32 layout: 1 block per ROI, 224 threads (7 waves), 1 thread per output
// pixel (196 active). Per-pixel corner indices + lerp weights live in
// registers and are reused across all 256 channels. Lanes map to (py,px) so
// stores coalesce to 128 B per wave and ~10 lanes share each gathered line.
//
// CDNA5 specifics used:
//  - global_load_async_to_lds_b32 + s_wait_asynccnt: one ASYNCcnt-tracked DMA
//    fetch of the 5-float ROI descriptor broadcast via LDS to all 7 waves
//    (instead of 7 per-wave scalar fetch clauses). Once per block, off the
//    hot path.
//  - readfirstlane re-uniforms the batch index so the gather loop keeps the
//    SGPR-base (SADDR+voffset) addressing form.
//  - __builtin_nontemporal_store -> global_store_b32 th:TH_STORE_NT.

namespace {

constexpr int kCropH = 14;
constexpr int kCropW = 14;
constexpr int kNPix  = kCropH * kCropW;   // 196
constexpr int kC     = 256;
constexpr int kH     = 128;
constexpr int kW     = 128;
constexpr int kHW    = kH * kW;
constexpr float kScale = 0.0625f;

__global__ __launch_bounds__(224) void roialign_nt_kernel(
    const float* __restrict__ feat,   // [B, C, H, W]
    const float* __restrict__ rois,   // [N, 5] = (id, x0, y0, x1, y1) image coords
    float* __restrict__ out) {        // [N, C, 14, 14]
  __shared__ float s_roi[8];

  const int n = blockIdx.x;
  const int t = threadIdx.x;

  // --- Fetch this block's ROI descriptor via the CDNA5 async DMA path. ---
  // Lanes 0..4 of wave 0 issue one async global->LDS copy each (no VGPR
  // return, tracked by ASYNCcnt), then the block synchronizes on the data.
  if (t < 5) {
    const float* g = rois + (size_t)n * 5 + t;
    // Low 32 bits of the flat shared-memory address == workgroup-relative
    // LDS byte offset (aperture form), which is what the instruction takes.
    unsigned lds_off = (unsigned)(size_t)&s_roi[t];
    asm volatile("global_load_async_to_lds_b32 %0, %1, off"
                 :: "v"(lds_off), "v"(g)
                 : "memory");
  }
  asm volatile("s_wait_asynccnt 0x0" ::: "memory");
  __syncthreads();

  if (t >= kNPix) return;             // lanes 196..223 idle after the barrier

  const float bid = s_roi[0];
  // Replicate the reference arithmetic exactly: normalize, then rescale.
  const float x0n = s_roi[1] * kScale / (float)(kW - 1);
  const float y0n = s_roi[2] * kScale / (float)(kH - 1);
  const float x1n = s_roi[3] * kScale / (float)(kW - 1);
  const float y1n = s_roi[4] * kScale / (float)(kH - 1);

  const float hsc = (y1n - y0n) * (float)(kH - 1) / (float)(kCropH - 1);
  const float wsc = (x1n - x0n) * (float)(kW - 1) / (float)(kCropW - 1);

  const int py = t / kCropW;
  const int px = t - py * kCropW;

  const float in_y = y0n * (float)(kH - 1) + (float)py * hsc;
  const float in_x = x0n * (float)(kW - 1) + (float)px * wsc;

  const bool valid = (in_y >= 0.0f) && (in_y <= (float)(kH - 1)) &&
                     (in_x >= 0.0f) && (in_x <= (float)(kW - 1));

  const float fy = floorf(in_y);
  const float fx = floorf(in_x);
  const float ly = in_y - fy;
  const float lx = in_x - fx;

  // Clamped corner indices: always in-bounds, so the hot loop is branch-free
  // and the invalid (extrapolated) case is a cheap select at the end.
  const int ty = (int)fminf(fmaxf(fy,        0.0f), (float)(kH - 1));
  const int by = (int)fminf(fmaxf(fy + 1.0f, 0.0f), (float)(kH - 1));
  const int xl = (int)fminf(fmaxf(fx,        0.0f), (float)(kW - 1));
  const int xr = (int)fminf(fmaxf(fx + 1.0f, 0.0f), (float)(kW - 1));

  const int s_tl = ty * kW + xl;
  const int s_tr = ty * kW + xr;
  const int s_bl = by * kW + xl;
  const int s_br = by * kW + xr;

  // Batch index is uniform across the block: pin it back into an SGPR so the
  // gather base stays scalar (SADDR form) and per-lane work is just voffsets.
  const int b = __builtin_amdgcn_readfirstlane((int)bid);
  const float* __restrict__ fb = feat + (size_t)b * (size_t)(kC * kHW);
  float* __restrict__ op = out + (size_t)n * (size_t)(kC * kNPix) + t;

  // 256 channel iterations; unroll 8 -> 32 in-flight L2 gathers per thread
  // (LOADcnt is 6 bits / 63 max) to cover latency even at low residency.
#pragma unroll 8
  for (int c = 0; c < kC; ++c) {
    const float* __restrict__ f = fb + c * kHW;
    const float tl = f[s_tl];
    const float tr = f[s_tr];
    const float bl = f[s_bl];
    const float br = f[s_br];
    const float top = tl + (tr - tl) * lx;
    const float bot = bl + (br - bl) * lx;
    float v = top + (bot - top) * ly;
    v = valid ? v : 0.0f;
    // Streaming output: non-temporal store keeps features L2-resident.
    __builtin_nontemporal_store(v, op + c * kNPix);
  }
}

}  // namespace

extern "C" void kernel_launch(void* const* d_in, const int* in_sizes, int n_in,
                              void* d_out, int out_size, void* d_ws, size_t ws_size,
                              hipStream_t stream) {
  const float* feat = (const float*)d_in[0];   // [4, 256, 128, 128] f32
  const float* rois = (const float*)d_in[1];   // [N, 5] f32
  float* out = (float*)d_out;                  // [N, 256, 14, 14] f32
  const int n_rois = in_sizes[1] / 5;
  if (n_rois <= 0) return;
  roialign_nt_kernel<<<dim3(n_rois), dim3(224), 0, stream>>>(feat, rois, out);
}